// _ITreeLSTMCell_39273180954986
// MI455X (gfx1250) — compile-verified
//
#include <hip/hip_runtime.h>
#include <hip/hip_bf16.h>

// Child-Sum TreeLSTM cell on gfx1250 (MI455X), fp32 WMMA (v_wmma_f32_16x16x4_f32).
//
// Pipeline (all on `stream`):
//  1) init:   zero hsum[N,128] (ws) and the c-half of d_out (fc accumulator)
//  2) projf:  xf = x @ w_f^T + b_f                       (node GEMM, WMMA)
//  3) edge:   hsum[dst] += h[src]           (atomics)
//             f = sigmoid(xf[dst] + h[src]@u_f^T)        (edge GEMM, WMMA)
//             c_acc[dst] += f * c[src]      (atomics)
//  4) final:  iou = [x|hsum] @ [w_iou|u_iou]^T + b_iou   (K=256 GEMM, WMMA)
//             c = sig(i)*tanh(u) + c_acc ; h = sig(o)*tanh(c)
//
// Uses segment_sum(h_src @ U^T) == segment_sum(h_src) @ U^T to turn the big
// edge GEMM into a node GEMM (halves FLOPs, quarters atomic traffic).

#define HS 128

typedef float v2f __attribute__((ext_vector_type(2)));
typedef float v8f __attribute__((ext_vector_type(8)));

__device__ __forceinline__ float sigmoidf_(float v) {
    return 1.0f / (1.0f + __expf(-v));
}

__device__ __forceinline__ v8f wmma_f32(v2f a, v2f b, v8f c) {
    // D = A(16x4,f32) * B(4x16,f32) + C(16x16,f32)
    return __builtin_amdgcn_wmma_f32_16x16x4_f32(false, a, false, b, (short)0, c,
                                                 false, false);
}

// ---------------------------------------------------------------- init -----
__global__ void _tl_init_kernel(float* __restrict__ hsum,
                                float* __restrict__ c_acc, int n4) {
    int i = blockIdx.x * blockDim.x + threadIdx.x;
    if (i < n4) {
        float4 z = make_float4(0.f, 0.f, 0.f, 0.f);
        reinterpret_cast<float4*>(hsum)[i]  = z;
        reinterpret_cast<float4*>(c_acc)[i] = z;
    }
}

// ------------------------------------------------- xf = x @ w_f^T + b_f ----
__global__ void __launch_bounds__(256)
_tl_projf_kernel(const float* __restrict__ x, const float* __restrict__ w_f,
                 const float* __restrict__ b_f, float* __restrict__ xf,
                 int n_tiles) {
    const int wave = (blockIdx.x * blockDim.x + threadIdx.x) >> 5;
    const int lane = threadIdx.x & 31;
    if (wave >= n_tiles) return;               // wave-uniform guard
    const int half = lane >> 4;                // 0: K={0,1}, 1: K={2,3}
    const int lm   = lane & 15;                // A: M, B: N, D: N
    const int rb   = wave * 16;

    v8f acc[8];
#pragma unroll
    for (int t = 0; t < 8; ++t) acc[t] = (v8f){0.f,0.f,0.f,0.f,0.f,0.f,0.f,0.f};

    const float* arow = x + (size_t)(rb + lm) * HS + 2 * half;
    for (int k0 = 0; k0 < HS; k0 += 4) {
        v2f a = *reinterpret_cast<const v2f*>(arow + k0);
#pragma unroll
        for (int nt = 0; nt < 8; ++nt) {
            v2f b = *reinterpret_cast<const v2f*>(w_f + (size_t)(nt * 16 + lm) * HS +
                                                  k0 + 2 * half);
            acc[nt] = wmma_f32(a, b, acc[nt]);
        }
    }
#pragma unroll
    for (int nt = 0; nt < 8; ++nt) {
        float bias = b_f[nt * 16 + lm];
#pragma unroll
        for (int r = 0; r < 8; ++r) {
            int row = rb + r + 8 * half;
            xf[(size_t)row * HS + nt * 16 + lm] = acc[nt][r] + bias;
        }
    }
}

// --------------------------------------------------------------- edges -----
__global__ void __launch_bounds__(256)
_tl_edge_kernel(const float* __restrict__ h_child,
                const float* __restrict__ c_child,
                const float* __restrict__ u_f, const float* __restrict__ xf,
                const int* __restrict__ esrc, const int* __restrict__ edst,
                float* __restrict__ hsum, float* __restrict__ c_acc,
                int n_tiles) {
    const int wave = (blockIdx.x * blockDim.x + threadIdx.x) >> 5;
    const int lane = threadIdx.x & 31;
    if (wave >= n_tiles) return;               // wave-uniform guard
    const int half = lane >> 4;
    const int lm   = lane & 15;
    const int e0   = wave * 16;

    // ---- uf = h_src @ u_f^T for 16 edges (WMMA) ----
    const int s_lm = esrc[e0 + lm];
    v8f acc[8];
#pragma unroll
    for (int t = 0; t < 8; ++t) acc[t] = (v8f){0.f,0.f,0.f,0.f,0.f,0.f,0.f,0.f};

    const float* arow = h_child + (size_t)s_lm * HS + 2 * half;
    for (int k0 = 0; k0 < HS; k0 += 4) {
        v2f a = *reinterpret_cast<const v2f*>(arow + k0);
#pragma unroll
        for (int nt = 0; nt < 8; ++nt) {
            v2f b = *reinterpret_cast<const v2f*>(u_f + (size_t)(nt * 16 + lm) * HS +
                                                  k0 + 2 * half);
            acc[nt] = wmma_f32(a, b, acc[nt]);
        }
    }

    // ---- hsum[dst] += h_child[src] (lane: edge lm, channel half) ----
    {
        const int d_lm = edst[e0 + lm];
        const float* hrow = h_child + (size_t)s_lm * HS;
        float* hsrow      = hsum + (size_t)d_lm * HS;
        const int c0 = half * 64;
#pragma unroll 8
        for (int c = c0; c < c0 + 64; ++c)
            unsafeAtomicAdd(&hsrow[c], hrow[c]);
    }

    // ---- f = sigmoid(xf[dst] + uf); c_acc[dst] += f * c_child[src] ----
#pragma unroll
    for (int r = 0; r < 8; ++r) {
        const int e  = e0 + r + 8 * half;      // edge index held in acc[.][r]
        const int sr = esrc[e];
        const int dr = edst[e];
        const float* crow  = c_child + (size_t)sr * HS;
        const float* xfrow = xf + (size_t)dr * HS;
        float* carow       = c_acc + (size_t)dr * HS;
#pragma unroll
        for (int nt = 0; nt < 8; ++nt) {
            int n = nt * 16 + lm;
            float f = sigmoidf_(xfrow[n] + acc[nt][r]);
            unsafeAtomicAdd(&carow[n], f * crow[n]);
        }
    }
}

// -------------------------------------------- final node update (K=256) ----
__global__ void __launch_bounds__(256, 1)
_tl_final_kernel(const float* __restrict__ x, const float* __restrict__ hsum,
                 const float* __restrict__ w_iou,
                 const float* __restrict__ u_iou,
                 const float* __restrict__ b_iou, float* __restrict__ h_out,
                 float* __restrict__ c_out, int n_tiles) {
    const int wave = (blockIdx.x * blockDim.x + threadIdx.x) >> 5;
    const int lane = threadIdx.x & 31;
    if (wave >= n_tiles) return;               // wave-uniform guard
    const int half = lane >> 4;
    const int lm   = lane & 15;
    const int rb   = wave * 16;

    v8f acc[24];                               // 24 N-tiles: i(8) o(8) u(8)
#pragma unroll
    for (int t = 0; t < 24; ++t)
        acc[t] = (v8f){0.f,0.f,0.f,0.f,0.f,0.f,0.f,0.f};

    // pass 1: x @ w_iou^T
    {
        const float* arow = x + (size_t)(rb + lm) * HS + 2 * half;
        for (int k0 = 0; k0 < HS; k0 += 4) {
            v2f a = *reinterpret_cast<const v2f*>(arow + k0);
#pragma unroll
            for (int nt = 0; nt < 24; ++nt) {
                v2f b = *reinterpret_cast<const v2f*>(
                    w_iou + (size_t)(nt * 16 + lm) * HS + k0 + 2 * half);
                acc[nt] = wmma_f32(a, b, acc[nt]);
            }
        }
    }
    // pass 2: hsum @ u_iou^T
    {
        const float* arow = hsum + (size_t)(rb + lm) * HS + 2 * half;
        for (int k0 = 0; k0 < HS; k0 += 4) {
            v2f a = *reinterpret_cast<const v2f*>(arow + k0);
#pragma unroll
            for (int nt = 0; nt < 24; ++nt) {
                v2f b = *reinterpret_cast<const v2f*>(
                    u_iou + (size_t)(nt * 16 + lm) * HS + k0 + 2 * half);
                acc[nt] = wmma_f32(a, b, acc[nt]);
            }
        }
    }

    // epilogue: gates + cell/hidden state
#pragma unroll
    for (int r = 0; r < 8; ++r) {
        const size_t row = (size_t)(rb + r + 8 * half);
#pragma unroll
        for (int nt = 0; nt < 8; ++nt) {
            int n = nt * 16 + lm;
            float iv = sigmoidf_(acc[nt][r]      + b_iou[n]);
            float ov = sigmoidf_(acc[8 + nt][r]  + b_iou[HS + n]);
            float uv = tanhf(acc[16 + nt][r]     + b_iou[2 * HS + n]);
            float cc = iv * uv + c_out[row * HS + n];   // + fc_sum
            c_out[row * HS + n] = cc;
            h_out[row * HS + n] = ov * tanhf(cc);
        }
    }
}

// -------------------------------------------------------------- launch -----
extern "C" void kernel_launch(void* const* d_in, const int* in_sizes, int n_in,
                              void* d_out, int out_size, void* d_ws,
                              size_t ws_size, hipStream_t stream) {
    const float* x       = (const float*)d_in[0];
    const float* h_child = (const float*)d_in[1];
    const float* c_child = (const float*)d_in[2];
    const float* w_iou   = (const float*)d_in[3];
    const float* b_iou   = (const float*)d_in[4];
    const float* w_f     = (const float*)d_in[5];
    const float* b_f     = (const float*)d_in[6];
    const float* u_iou   = (const float*)d_in[7];
    const float* u_f     = (const float*)d_in[8];
    const int*   esrc    = (const int*)d_in[9];
    const int*   edst    = (const int*)d_in[10];

    const int N = in_sizes[0] / HS;
    const int E = in_sizes[9];

    float* h_out = (float*)d_out;
    float* c_out = h_out + (size_t)N * HS;     // also fc_sum accumulator

    float* xf   = (float*)d_ws;                // [N,128]
    float* hsum = xf + (size_t)N * HS;         // [N,128]

    // 1) zero accumulators
    {
        int n4 = (N * HS) / 4;
        _tl_init_kernel<<<(n4 + 255) / 256, 256, 0, stream>>>(hsum, c_out, n4);
    }
    // 2) xf projection  (N divisible by 16 for given sizes)
    const int node_tiles = N / 16;
    const int node_blocks = (node_tiles + 7) / 8;   // 8 waves per 256-thr block
    _tl_projf_kernel<<<node_blocks, 256, 0, stream>>>(x, w_f, b_f, xf,
                                                      node_tiles);
    // 3) edge pass
    const int e_tiles = E / 16;
    _tl_edge_kernel<<<(e_tiles + 7) / 8, 256, 0, stream>>>(
        h_child, c_child, u_f, xf, esrc, edst, hsum, c_out, e_tiles);
    // 4) final node update
    _tl_final_kernel<<<node_blocks, 256, 0, stream>>>(
        x, hsum, w_iou, u_iou, b_iou, h_out, c_out, node_tiles);
}